// GraphConvolution_8323646620047
// MI455X (gfx1250) — compile-verified
//
#include <hip/hip_runtime.h>

typedef __attribute__((ext_vector_type(2))) float v2f;
typedef __attribute__((ext_vector_type(8))) float v8f;

#define N_NODES 100000
#define N_EDGES 3200000
#define DIM     256

// ---------------------------------------------------------------------------
// Kernel 1: out[n, :] = b[:]   (d_out is poisoned; bias folded into init)
// ---------------------------------------------------------------------------
__global__ __launch_bounds__(256) void gcn_bias_init(const float* __restrict__ b,
                                                     float* __restrict__ out) {
  const size_t i = (size_t)blockIdx.x * blockDim.x + threadIdx.x;   // float4 units
  const size_t total4 = (size_t)N_NODES * (DIM / 4);
  if (i < total4) {
    const float4* b4 = (const float4*)b;
    ((float4*)out)[i] = b4[i & (DIM / 4 - 1)];
  }
}

// ---------------------------------------------------------------------------
// Kernel 2: support = x @ W via V_WMMA_F32_16X16X4_F32 (full fp32 precision).
// One wave computes a 16x64 output strip: 4 independent 16x16 accumulators so
// consecutive WMMAs have no D->C RAW chain (XDL pipelines instead of
// serializing), and the A fragment is loaded once per K-step and reused 4x.
//
// A (16x4) lane layout (ISA 7.12.2): lanes 0-15 M=0..15 hold K=k0+0/k0+1 in
// v[0]/v[1]; lanes 16-31 hold K=k0+2/k0+3  -> per-lane contiguous float2.
// B (4x16): rows striped across lanes (mirror of A): lane l holds column
// n0+(l&15), rows k0+2*(l>>4) and k0+2*(l>>4)+1.
// C/D (16x16): VGPR r, lanes 0-15 -> M=r, lanes 16-31 -> M=r+8, N = lane&15.
// ---------------------------------------------------------------------------
__global__ __launch_bounds__(256) void gcn_gemm_wmma(const float* __restrict__ x,
                                                     const float* __restrict__ w,
                                                     float* __restrict__ support) {
  const int lane = threadIdx.x & 31;
  const int wave = threadIdx.x >> 5;
  const int id   = blockIdx.x * 8 + wave;        // 25000 wave-jobs, exact
  const int mt   = id >> 2;                      // 6250 M-tiles (shared in block)
  const int ng   = id & 3;                       // 4 N-groups of 64 columns
  const int m0   = mt << 4;
  const int n0   = ng << 6;
  const int l15  = lane & 15;
  const int khalf = (lane >> 4) << 1;            // 0 for lanes 0-15, 2 for 16-31

  const float* xrow  = x + (size_t)(m0 + l15) * DIM + khalf;   // A source
  const float* wbase = w + (size_t)khalf * DIM + (n0 + l15);   // B source

  v8f acc0 = {}, acc1 = {}, acc2 = {}, acc3 = {};
#pragma unroll 4
  for (int k0 = 0; k0 < DIM; k0 += 4) {
    v2f a = *(const v2f*)(xrow + k0);            // global_load_b64, reused 4x
    const float* wk = wbase + (size_t)k0 * DIM;
    v2f b0; b0.x = wk[0];  b0.y = wk[DIM];       // W[k][n0+ 0+l15], W[k+1][..]
    v2f b1; b1.x = wk[16]; b1.y = wk[DIM + 16];
    v2f b2; b2.x = wk[32]; b2.y = wk[DIM + 32];
    v2f b3; b3.x = wk[48]; b3.y = wk[DIM + 48];
    acc0 = __builtin_amdgcn_wmma_f32_16x16x4_f32(false, a, false, b0, (short)0, acc0, false, false);
    acc1 = __builtin_amdgcn_wmma_f32_16x16x4_f32(false, a, false, b1, (short)0, acc1, false, false);
    acc2 = __builtin_amdgcn_wmma_f32_16x16x4_f32(false, a, false, b2, (short)0, acc2, false, false);
    acc3 = __builtin_amdgcn_wmma_f32_16x16x4_f32(false, a, false, b3, (short)0, acc3, false, false);
  }

  float* cbase = support + (size_t)(m0 + (lane >> 4) * 8) * DIM + (n0 + l15);
#pragma unroll
  for (int r = 0; r < 8; ++r) {
    cbase[(size_t)r * DIM +  0] = acc0[r];
    cbase[(size_t)r * DIM + 16] = acc1[r];
    cbase[(size_t)r * DIM + 32] = acc2[r];
    cbase[(size_t)r * DIM + 48] = acc3[r];
  }
}

// ---------------------------------------------------------------------------
// Kernel 3: one wave per edge. Gather 1KB row of support (2x b128 per lane,
// L2-resident), scale by edge_val, scatter with 8 no-return f32 atomics/lane.
// Edge index/value arrays are streamed exactly once -> non-temporal loads so
// they don't evict the hot support/out working set from the 192MB L2.
// ---------------------------------------------------------------------------
__global__ __launch_bounds__(256) void gcn_scatter(const float* __restrict__ support,
                                                   const int* __restrict__ erow,
                                                   const int* __restrict__ ecol,
                                                   const float* __restrict__ eval,
                                                   float* __restrict__ out) {
  const int lane = threadIdx.x & 31;
  const int e    = (int)((blockIdx.x * blockDim.x + threadIdx.x) >> 5);
  if (e >= N_EDGES) return;

  const int   r = __builtin_nontemporal_load(erow + e);
  const int   c = __builtin_nontemporal_load(ecol + e);
  const float v = __builtin_nontemporal_load(eval + e);

  const float4* src = (const float4*)(support + (size_t)c * DIM);
  float*        dst = out + (size_t)r * DIM;

  float4 s0 = src[lane];        // elements [lane*4 .. lane*4+3]
  float4 s1 = src[lane + 32];   // elements [128+lane*4 .. 128+lane*4+3]

  const int d0 = lane * 4;
  unsafeAtomicAdd(dst + d0 + 0,       s0.x * v);
  unsafeAtomicAdd(dst + d0 + 1,       s0.y * v);
  unsafeAtomicAdd(dst + d0 + 2,       s0.z * v);
  unsafeAtomicAdd(dst + d0 + 3,       s0.w * v);
  unsafeAtomicAdd(dst + d0 + 128,     s1.x * v);
  unsafeAtomicAdd(dst + d0 + 128 + 1, s1.y * v);
  unsafeAtomicAdd(dst + d0 + 128 + 2, s1.z * v);
  unsafeAtomicAdd(dst + d0 + 128 + 3, s1.w * v);
}

// ---------------------------------------------------------------------------
extern "C" void kernel_launch(void* const* d_in, const int* in_sizes, int n_in,
                              void* d_out, int out_size, void* d_ws, size_t ws_size,
                              hipStream_t stream) {
  const float* x    = (const float*)d_in[0];   // [N_NODES, DIM]
  const int*   erow = (const int*)  d_in[1];   // [N_EDGES]
  const int*   ecol = (const int*)  d_in[2];   // [N_EDGES]
  const float* eval = (const float*)d_in[3];   // [N_EDGES]
  const float* w    = (const float*)d_in[4];   // [DIM, DIM]
  const float* b    = (const float*)d_in[5];   // [DIM]
  float* out     = (float*)d_out;              // [N_NODES, DIM]
  float* support = (float*)d_ws;               // needs N_NODES*DIM*4 = 102.4 MB

  // 1) out = broadcast(bias)
  {
    const int total4 = N_NODES * (DIM / 4);            // 6.4M float4
    gcn_bias_init<<<(total4 + 255) / 256, 256, 0, stream>>>(b, out);
  }
  // 2) support = x @ W  (WMMA fp32, 16x64 strip per wave)
  {
    const int jobs = (N_NODES / 16) * (DIM / 64);      // 25000 wave-jobs
    gcn_gemm_wmma<<<jobs / 8, 256, 0, stream>>>(x, w, support);
  }
  // 3) scatter-add edges (8 waves/block, one edge per wave)
  {
    gcn_scatter<<<N_EDGES / 8, 256, 0, stream>>>(support, erow, ecol, eval, out);
  }
}